// DCT2D_90907277787227
// MI455X (gfx1250) — compile-verified
//
#include <hip/hip_runtime.h>

// CDNA5 / gfx1250: wave32, WMMA f32 16x16x4.
typedef float v2f __attribute__((ext_vector_type(2)));
typedef float v4f __attribute__((ext_vector_type(4)));
typedef float v8f __attribute__((ext_vector_type(8)));

#define PI_F 3.14159265358979323846f

// Dblk = diag(D, D), D = orthonormal 8x8 DCT-II matrix.
// D[a][n] = s(a) * cos(pi*(2n+1)*a/16), s(0)=sqrt(1/8), else sqrt(2/8)=0.5
__device__ __forceinline__ float dblk(int i, int j) {
    if ((i >> 3) != (j >> 3)) return 0.0f;
    int a = i & 7, n = j & 7;
    float s = (a == 0) ? 0.35355339059327373f : 0.5f;
    return s * __cosf(PI_F * (2.0f * (float)n + 1.0f) * (float)a * (1.0f / 16.0f));
}

// Input : x  (8, 3, 256, 256) f32
// Output: out(8, 192, 256, 256) f32  (64 DCT coeffs -> channels, 8x8 replicated)
//
// One wave handles one 16x16 tile (= 2x2 grid of 8x8 DCT blocks).
// Tile grid: 16x16 tiles per (b,c) image -> 8*3*16*16 = 6144 tiles.
__global__ __launch_bounds__(256) void dct8x8_wmma_kernel(
    const float* __restrict__ x, float* __restrict__ out) {

    const int lane = threadIdx.x & 31;
    const int wave = threadIdx.x >> 5;
    const int g    = blockIdx.x * 8 + wave;     // tile id, [0, 6144)

    const int tj = g & 15;                      // tile col in 16-tile grid
    const int ti = (g >> 4) & 15;               // tile row
    const int bc = g >> 8;                      // b*3 + c
    const int c  = bc % 3;
    const int b  = bc / 3;

    const int half = lane >> 4;                 // 0: lanes 0-15, 1: lanes 16-31
    const int lr   = lane & 15;

    // ---- constant A operand: Dblk columns per K-step, A-layout ----
    // step k: VGPR0 = Dblk[row][4k + 2*half], VGPR1 = Dblk[row][4k + 2*half + 1]
    v2f amat[4];
#pragma unroll
    for (int k = 0; k < 4; ++k) {
        int cc = 4 * k + 2 * half;
        amat[k].x = dblk(lr, cc);
        amat[k].y = dblk(lr, cc + 1);
    }

    // ---- stage 1: U = Dblk * X ; X loaded from global directly in B-layout ----
    // step k B operand: VGPR0 = X[4k + 2*half][lr], VGPR1 = X[4k + 2*half + 1][lr]
    const float* xt = x + (((b * 3 + c) * 256 + ti * 16) * 256 + tj * 16);
    v8f u = {};
#pragma unroll
    for (int k = 0; k < 4; ++k) {
        int r0 = 4 * k + 2 * half;
        v2f bv;
        bv.x = xt[(r0    ) * 256 + lr];
        bv.y = xt[(r0 + 1) * 256 + lr];
        u = __builtin_amdgcn_wmma_f32_16x16x4_f32(false, amat[k], false, bv,
                                                  (short)0, u, false, false);
    }

    // ---- transpose U via per-wave LDS slice (stride 17 kills bank conflicts) ----
    __shared__ float lds[8][16 * 17];
    float* T = &lds[wave][0];
#pragma unroll
    for (int v = 0; v < 8; ++v) {
        int row = v + 8 * half;                 // D-layout: lane holds U[row][lr]
        T[row * 17 + lr] = u[v];
    }
    asm volatile("s_wait_dscnt 0" ::: "memory");

    // ---- stage 2: V = Dblk * U^T = (Dblk * X * Dblk^T)^T = Y^T ----
    v8f y = {};
#pragma unroll
    for (int k = 0; k < 4; ++k) {
        int cc = 4 * k + 2 * half;              // row of U^T == column of U
        v2f bv;
        bv.x = T[lr * 17 + cc];
        bv.y = T[lr * 17 + cc + 1];
        y = __builtin_amdgcn_wmma_f32_16x16x4_f32(false, amat[k], false, bv,
                                                  (short)0, y, false, false);
    }

    // ---- scatter: each lane holds V[v + 8*half][lr] = Y[lr][v + 8*half] ----
    // Y row p = lr  -> pi = lr>>3 (block row in tile), a = lr&7 (row freq)
    // Y col q = v+8*half -> qj = half (block col in tile), d = v (col freq)
    const int pi = lr >> 3, af = lr & 7;
    const int bi = 2 * ti + pi;                 // 8x8 block row in [0,32)
    const int bj = 2 * tj + half;               // 8x8 block col in [0,32)

#pragma unroll
    for (int v = 0; v < 8; ++v) {
        float val = y[v];
        int ch = c * 64 + af * 8 + v;           // output channel
        float* o = out + (((b * 192 + ch) * 256 + bi * 8) * 256 + bj * 8);
        v4f vv = {val, val, val, val};
        // 403 MB of output > 192 MB L2: stream with non-temporal stores.
#pragma unroll
        for (int r = 0; r < 8; ++r) {
            __builtin_nontemporal_store(vv, (v4f*)(o + r * 256));
            __builtin_nontemporal_store(vv, (v4f*)(o + r * 256 + 4));
        }
    }
}

extern "C" void kernel_launch(void* const* d_in, const int* in_sizes, int n_in,
                              void* d_out, int out_size, void* d_ws, size_t ws_size,
                              hipStream_t stream) {
    const float* x = (const float*)d_in[0];
    float* out = (float*)d_out;
    // 8*3*16*16 = 6144 tiles, 8 waves (256 threads) per block -> 768 blocks
    dct8x8_wmma_kernel<<<dim3(768), dim3(256), 0, stream>>>(x, out);
}